// SentenceDecoder_41051297415883
// MI455X (gfx1250) — compile-verified
//
#include <hip/hip_runtime.h>
#include <hip/hip_bf16.h>

#define HDIM 1024
#define TK   1024
#define BATCH 64

typedef __attribute__((ext_vector_type(2))) float v2f;
typedef __attribute__((ext_vector_type(8))) float v8f;

// gate stride inside W_ih / W_hh: HDIM rows of HDIM floats = 1,048,576 floats (4 MB)
#define GSTRIDE (HDIM * HDIM)

__device__ __forceinline__ float sigmoidf_(float x) {
    return 1.0f / (1.0f + __expf(-x));
}

__device__ __forceinline__ v2f ldv2(const float* p) {
    const float2 t = *(const float2*)p;
    return v2f{ t.x, t.y };
}

// ---------------------------------------------------------------------------
// Kernel 1: fused LSTM cell.
// One wave per 16x16 (batch x hidden) tile; i/f/g/o accumulated in 4 v8f regs.
// Per-lane base pointers advanced by +4 floats per K-step; gate offsets are
// compile-time immediates (0 / 4MB; second base covers 8MB / 12MB).
// grid = (H/16, B/16) = (64, 4).
// ---------------------------------------------------------------------------
__global__ __launch_bounds__(32)
void lstm_wmma_kernel(const float* __restrict__ x,    // [B,H] w_hx
                      const float* __restrict__ hprev,// [B,H] s_hx
                      const float* __restrict__ cprev,// [B,H] s_cx
                      const float* __restrict__ Wih,  // [4H,H]
                      const float* __restrict__ Whh,  // [4H,H]
                      const float* __restrict__ bih,  // [4H]
                      const float* __restrict__ bhh,  // [4H]
                      float* __restrict__ h_new,      // [B,H]
                      float* __restrict__ c_new)      // [B,H]
{
    const int lane = threadIdx.x & 31;
    const int l    = lane & 15;
    const int half = lane >> 4;
    const int n0   = blockIdx.x * 16;   // hidden tile
    const int m0   = blockIdx.y * 16;   // batch tile

    v8f acc0 = {}, acc1 = {}, acc2 = {}, acc3 = {};

    // per-lane streaming pointers (advance +4 floats per K step)
    const float* px   = x     + (m0 + l) * HDIM + 2 * half;
    const float* ph   = hprev + (m0 + l) * HDIM + 2 * half;
    const float* pwiA = Wih   + (n0 + l) * HDIM + 2 * half;   // gates i (+0), f (+4MB)
    const float* pwiB = pwiA  + 2 * GSTRIDE;                  // gates g (+0), o (+4MB)
    const float* pwhA = Whh   + (n0 + l) * HDIM + 2 * half;
    const float* pwhB = pwhA  + 2 * GSTRIDE;

#pragma unroll 4
    for (int k0 = 0; k0 < HDIM; k0 += 4) {
        v2f ax = ldv2(px);
        v2f ah = ldv2(ph);

        v2f bi0 = ldv2(pwiA);             // W_ih gate i
        v2f bi1 = ldv2(pwiA + GSTRIDE);   // W_ih gate f
        v2f bi2 = ldv2(pwiB);             // W_ih gate g
        v2f bi3 = ldv2(pwiB + GSTRIDE);   // W_ih gate o
        v2f bh0 = ldv2(pwhA);
        v2f bh1 = ldv2(pwhA + GSTRIDE);
        v2f bh2 = ldv2(pwhB);
        v2f bh3 = ldv2(pwhB + GSTRIDE);

        acc0 = __builtin_amdgcn_wmma_f32_16x16x4_f32(false, ax, false, bi0, (short)0, acc0, false, false);
        acc0 = __builtin_amdgcn_wmma_f32_16x16x4_f32(false, ah, false, bh0, (short)0, acc0, false, false);
        acc1 = __builtin_amdgcn_wmma_f32_16x16x4_f32(false, ax, false, bi1, (short)0, acc1, false, false);
        acc1 = __builtin_amdgcn_wmma_f32_16x16x4_f32(false, ah, false, bh1, (short)0, acc1, false, false);
        acc2 = __builtin_amdgcn_wmma_f32_16x16x4_f32(false, ax, false, bi2, (short)0, acc2, false, false);
        acc2 = __builtin_amdgcn_wmma_f32_16x16x4_f32(false, ah, false, bh2, (short)0, acc2, false, false);
        acc3 = __builtin_amdgcn_wmma_f32_16x16x4_f32(false, ax, false, bi3, (short)0, acc3, false, false);
        acc3 = __builtin_amdgcn_wmma_f32_16x16x4_f32(false, ah, false, bh3, (short)0, acc3, false, false);

        px += 4; ph += 4; pwiA += 4; pwiB += 4; pwhA += 4; pwhB += 4;
    }

    const int j = n0 + l;   // hidden index (per lane)
    const float bi_i = bih[j]            + bhh[j];
    const float bi_f = bih[j + HDIM]     + bhh[j + HDIM];
    const float bi_g = bih[j + 2 * HDIM] + bhh[j + 2 * HDIM];
    const float bi_o = bih[j + 3 * HDIM] + bhh[j + 3 * HDIM];

#pragma unroll
    for (int i = 0; i < 8; ++i) {
        const int m = m0 + i + 8 * half;        // batch row (C/D layout)
        const float gi = sigmoidf_(acc0[i] + bi_i);
        const float gf = sigmoidf_(acc1[i] + bi_f);
        const float gg = tanhf    (acc2[i] + bi_g);
        const float go = sigmoidf_(acc3[i] + bi_o);
        const float cn = gf * cprev[m * HDIM + j] + gi * gg;
        c_new[m * HDIM + j] = cn;
        h_new[m * HDIM + j] = go * tanhf(cn);
    }
}

// ---------------------------------------------------------------------------
// Kernel 2: dec_feature = h_new @ W_s^T + b_s     [B,H]
// ---------------------------------------------------------------------------
__global__ __launch_bounds__(32)
void decfeat_wmma_kernel(const float* __restrict__ hn,  // [B,H]
                         const float* __restrict__ Ws,  // [H,H]
                         const float* __restrict__ bs,  // [H]
                         float* __restrict__ df)        // [B,H]
{
    const int lane = threadIdx.x & 31;
    const int l    = lane & 15;
    const int half = lane >> 4;
    const int n0   = blockIdx.x * 16;
    const int m0   = blockIdx.y * 16;

    v8f acc = {};
    const float* pa = hn + (m0 + l) * HDIM + 2 * half;
    const float* pb = Ws + (n0 + l) * HDIM + 2 * half;

#pragma unroll 8
    for (int k0 = 0; k0 < HDIM; k0 += 4) {
        v2f a = ldv2(pa);
        v2f b = ldv2(pb);
        acc = __builtin_amdgcn_wmma_f32_16x16x4_f32(false, a, false, b, (short)0, acc, false, false);
        pa += 4; pb += 4;
    }
    const int j = n0 + l;
#pragma unroll
    for (int i = 0; i < 8; ++i) {
        const int m = m0 + i + 8 * half;
        df[m * HDIM + j] = acc[i] + bs[j];
    }
}

// ---------------------------------------------------------------------------
// Kernel 3: scores[t,b] = sum_h tanh(ef[t,b,h]+df[b,h]) * v_w[h] + v_b
// One 256-thread block per (t,b); float4 streaming (268 MB pass).
// Wave32 shuffle reduction + 8-entry LDS combine (1 barrier).
// ---------------------------------------------------------------------------
__global__ __launch_bounds__(256)
void scores_kernel(const float* __restrict__ ef,    // [T,B,H]
                   const float* __restrict__ df,    // [B,H]
                   const float* __restrict__ vw,    // [H]
                   const float* __restrict__ vb,    // [1]
                   float* __restrict__ scores)      // [T,B]
{
    const int tb  = blockIdx.x;
    const int b   = tb & (BATCH - 1);
    const int tid = threadIdx.x;

    const float4 e = ((const float4*)(ef + (size_t)tb * HDIM))[tid];
    const float4 d = ((const float4*)(df + (size_t)b * HDIM))[tid];
    const float4 v = ((const float4*)vw)[tid];

    float p = tanhf(e.x + d.x) * v.x + tanhf(e.y + d.y) * v.y +
              tanhf(e.z + d.z) * v.z + tanhf(e.w + d.w) * v.w;

    // wave32 reduction
#pragma unroll
    for (int off = 16; off > 0; off >>= 1)
        p += __shfl_down(p, off, 32);

    __shared__ float red[8];
    if ((tid & 31) == 0) red[tid >> 5] = p;
    __syncthreads();
    if (tid == 0) {
        float s = 0.0f;
#pragma unroll
        for (int i = 0; i < 8; ++i) s += red[i];
        scores[tb] = s + vb[0];
    }
}

// ---------------------------------------------------------------------------
// Kernel 4: softmax over t (per batch column) * mask.  grid = B = 64.
// ---------------------------------------------------------------------------
__global__ __launch_bounds__(256)
void softmax_kernel(const float* __restrict__ scores, // [T,B]
                    const float* __restrict__ mask,   // [T,B,1]
                    float* __restrict__ attn)         // [T,B]
{
    const int b   = blockIdx.x;
    const int tid = threadIdx.x;
    __shared__ float red[256];

    float mx = -3.4e38f;
    for (int t = tid; t < TK; t += 256)
        mx = fmaxf(mx, scores[t * BATCH + b]);
    red[tid] = mx; __syncthreads();
#pragma unroll
    for (int s = 128; s > 0; s >>= 1) {
        if (tid < s) red[tid] = fmaxf(red[tid], red[tid + s]);
        __syncthreads();
    }
    mx = red[0];
    __syncthreads();

    float sum = 0.0f;
    for (int t = tid; t < TK; t += 256)
        sum += __expf(scores[t * BATCH + b] - mx);
    red[tid] = sum; __syncthreads();
#pragma unroll
    for (int s = 128; s > 0; s >>= 1) {
        if (tid < s) red[tid] += red[tid + s];
        __syncthreads();
    }
    const float inv = 1.0f / red[0];

    for (int t = tid; t < TK; t += 256)
        attn[t * BATCH + b] =
            __expf(scores[t * BATCH + b] - mx) * inv * mask[t * BATCH + b];
}

// ---------------------------------------------------------------------------
// Kernel 5: content[b,h] = sum_t attn[t,b] * eo[t,b,h]   (268 MB pass)
// ---------------------------------------------------------------------------
__global__ __launch_bounds__(256)
void content_kernel(const float* __restrict__ attn,  // [T,B]
                    const float* __restrict__ eo,    // [T,B,H]
                    float* __restrict__ content)     // [B,H]
{
    const int b   = blockIdx.x;
    const int h0  = blockIdx.y * 256;
    const int tid = threadIdx.x;

    float acc = 0.0f;
    for (int t = 0; t < TK; ++t) {
        const size_t base = ((size_t)t * BATCH + b) * HDIM + h0 + tid;
        if (t + 1 < TK)
            __builtin_prefetch(&eo[base + (size_t)BATCH * HDIM], 0, 0);
        acc += attn[t * BATCH + b] * eo[base];
    }
    content[b * HDIM + h0 + tid] = acc;
}

// ---------------------------------------------------------------------------
// Kernel 6: final = tanh([content, h_new] @ lin_w^T + lin_b)   [B,H], K=2H
// ---------------------------------------------------------------------------
__global__ __launch_bounds__(32)
void final_wmma_kernel(const float* __restrict__ content, // [B,H]
                       const float* __restrict__ hn,      // [B,H]
                       const float* __restrict__ linw,    // [H,2H]
                       const float* __restrict__ linb,    // [H]
                       float* __restrict__ out)           // [B,H]
{
    const int lane = threadIdx.x & 31;
    const int l    = lane & 15;
    const int half = lane >> 4;
    const int n0   = blockIdx.x * 16;
    const int m0   = blockIdx.y * 16;

    v8f acc = {};
    const float* pc = content + (m0 + l) * HDIM     + 2 * half;
    const float* ph = hn      + (m0 + l) * HDIM     + 2 * half;
    const float* pw = linw    + (n0 + l) * 2 * HDIM + 2 * half;

    // first half of K: A = content
#pragma unroll 8
    for (int k0 = 0; k0 < HDIM; k0 += 4) {
        v2f a = ldv2(pc);
        v2f b = ldv2(pw);
        acc = __builtin_amdgcn_wmma_f32_16x16x4_f32(false, a, false, b, (short)0, acc, false, false);
        pc += 4; pw += 4;
    }
    // second half of K: A = h_new
#pragma unroll 8
    for (int k0 = 0; k0 < HDIM; k0 += 4) {
        v2f a = ldv2(ph);
        v2f b = ldv2(pw);
        acc = __builtin_amdgcn_wmma_f32_16x16x4_f32(false, a, false, b, (short)0, acc, false, false);
        ph += 4; pw += 4;
    }

    const int j = n0 + l;
#pragma unroll
    for (int i = 0; i < 8; ++i) {
        const int m = m0 + i + 8 * half;
        out[m * HDIM + j] = tanhf(acc[i] + linb[j]);
    }
}

// ---------------------------------------------------------------------------
extern "C" void kernel_launch(void* const* d_in, const int* in_sizes, int n_in,
                              void* d_out, int out_size, void* d_ws, size_t ws_size,
                              hipStream_t stream) {
    (void)in_sizes; (void)n_in; (void)out_size; (void)ws_size;

    const float* w_hx  = (const float*)d_in[0];
    const float* s_hx  = (const float*)d_in[1];
    const float* s_cx  = (const float*)d_in[2];
    const float* eo    = (const float*)d_in[3];
    const float* ef    = (const float*)d_in[4];
    const float* mask  = (const float*)d_in[5];
    const float* W_ih  = (const float*)d_in[6];
    const float* W_hh  = (const float*)d_in[7];
    const float* b_ih  = (const float*)d_in[8];
    const float* b_hh  = (const float*)d_in[9];
    const float* W_s   = (const float*)d_in[10];
    const float* b_s   = (const float*)d_in[11];
    const float* v_w   = (const float*)d_in[12];
    const float* v_b   = (const float*)d_in[13];
    const float* lin_w = (const float*)d_in[14];
    const float* lin_b = (const float*)d_in[15];

    float* out     = (float*)d_out;                 // final_dist [B,H]
    float* h_new   = out + BATCH * HDIM;            // h_new      [B,H]
    float* c_new   = out + 2 * BATCH * HDIM;        // c_new      [B,H]

    float* ws      = (float*)d_ws;
    float* df      = ws;                            // [B,H]   64K floats
    float* scores  = ws + 1 * BATCH * HDIM;         // [T,B]   64K floats
    float* attn    = ws + 2 * BATCH * HDIM;         // [T,B]   64K floats
    float* content = ws + 3 * BATCH * HDIM;         // [B,H]   64K floats

    dim3 gtile(HDIM / 16, BATCH / 16);              // (64, 4)

    lstm_wmma_kernel<<<gtile, 32, 0, stream>>>(w_hx, s_hx, s_cx, W_ih, W_hh,
                                               b_ih, b_hh, h_new, c_new);

    decfeat_wmma_kernel<<<gtile, 32, 0, stream>>>(h_new, W_s, b_s, df);

    scores_kernel<<<TK * BATCH, 256, 0, stream>>>(ef, df, v_w, v_b, scores);

    softmax_kernel<<<BATCH, 256, 0, stream>>>(scores, mask, attn);

    content_kernel<<<dim3(BATCH, HDIM / 256), 256, 0, stream>>>(attn, eo, content);

    final_wmma_kernel<<<gtile, 32, 0, stream>>>(content, h_new, lin_w, lin_b, out);
}